// GraphSageLayer_67645734912963
// MI455X (gfx1250) — compile-verified
//
#include <hip/hip_runtime.h>
#include <math.h>

#define N_NODES 50000
#define N_EDGES 600000
#define DIMS    128
#define TWO_D   256
#define N_TILES (N_NODES / 16) /* 3125, exact */
#define EPSN    1e-12f

typedef __attribute__((ext_vector_type(2))) float v2f;
typedef __attribute__((ext_vector_type(8))) float v8f;

// ---------------- zero workspace (summed[N*D] + deg[N]) ----------------
__global__ void zero_kernel(float* __restrict__ p, int n) {
  int i = blockIdx.x * blockDim.x + threadIdx.x;
  if (i < n) p[i] = 0.0f;
}

// ---------------- repack W[128][256] into WMMA B-fragment order --------
// B[k][n] = W[n][k].  Fragment (j = out-tile 0..7, t = k-step 0..63):
//   b.x (VGPR0): lanes 0-15 -> K=4t+0, lanes 16-31 -> K=4t+2, N = 16j + (lane&15)
//   b.y (VGPR1): K = b.x's K + 1
// Stored as wpack[(j*64+t)*64 + lane*2 + {0,1}]  (128 KB, L2-resident)
__global__ void repack_w_kernel(const float* __restrict__ W, float* __restrict__ wpack) {
  int id = blockIdx.x * blockDim.x + threadIdx.x;
  if (id >= 8 * 64 * 32) return;
  int lane = id & 31;
  int t    = (id >> 5) & 63;
  int j    = id >> 11;
  int n    = (j << 4) + (lane & 15);
  int k0   = (t << 2) + ((lane >> 4) << 1);
  wpack[id * 2 + 0] = W[n * TWO_D + k0 + 0];
  wpack[id * 2 + 1] = W[n * TWO_D + k0 + 1];
}

// ---------------- edge gather + L2-resident atomic scatter -------------
__global__ __launch_bounds__(256) void scatter_kernel(
    const float* __restrict__ h, const int* __restrict__ src, const int* __restrict__ dst,
    float* __restrict__ summed, float* __restrict__ deg) {
  int gw   = (int)((blockIdx.x * blockDim.x + threadIdx.x) >> 5); // wave per edge
  int lane = threadIdx.x & 31;
  if (gw >= N_EDGES) return;
  int s = src[gw];
  int d = dst[gw];
  const float* hr = h + (size_t)s * DIMS;
  float*       sr = summed + (size_t)d * DIMS;
#pragma unroll
  for (int i = 0; i < 4; ++i) {
    int c = lane + (i << 5);
    atomicAdd(&sr[c], hr[c]); // 25.8 MB accumulator fits in 192 MB L2
  }
  if (lane == 0) atomicAdd(&deg[d], 1.0f);
}

// ---------------- node apply: concat(h,c) @ W^T via WMMA f32 -----------
__global__ __launch_bounds__(256, 1) void node_apply_kernel(
    const float* __restrict__ h, const float* __restrict__ snorm,
    const float* __restrict__ summed, const float* __restrict__ deg,
    const float* __restrict__ wpack, float* __restrict__ out) {
  const int lane = threadIdx.x & 31;
  const int wave = threadIdx.x >> 5;
  const int tile = blockIdx.x * (blockDim.x >> 5) + wave;
  if (tile >= N_TILES) return; // wave-uniform: EXEC stays all-1s for WMMA

  const int base  = tile << 4;
  const int arow  = lane & 15;          // A-matrix M row owned by this lane
  const int khalf = (lane >> 4) << 1;   // K offset 0 (lanes 0-15) / 2 (lanes 16-31)
  const int anode = base + arow;

  const float  invd = 1.0f / fmaxf(deg[anode], 1.0f);
  const float* hrow = h      + (size_t)anode * DIMS;
  const float* crow = summed + (size_t)anode * DIMS;
  const float* wl   = wpack + (size_t)lane * 2;

  v8f acc[8] = {};

  // K = 0..127 : h part of concat
  for (int t = 0; t < 32; ++t) {
    const int k0 = (t << 2) + khalf;
    v2f a = *(const v2f*)(hrow + k0);
#pragma unroll
    for (int j = 0; j < 8; ++j) {
      v2f b = *(const v2f*)(wl + (((j << 6) + t) << 6));
      acc[j] = __builtin_amdgcn_wmma_f32_16x16x4_f32(
          false, a, false, b, (short)0, acc[j], false, false);
    }
  }
  // K = 128..255 : c = summed * inv_deg, fused into the A-fragment load
  for (int t = 32; t < 64; ++t) {
    const int k0 = (t << 2) + khalf - DIMS;
    v2f a = *(const v2f*)(crow + k0);
    a.x *= invd; a.y *= invd;
#pragma unroll
    for (int j = 0; j < 8; ++j) {
      v2f b = *(const v2f*)(wl + (((j << 6) + t) << 6));
      acc[j] = __builtin_amdgcn_wmma_f32_16x16x4_f32(
          false, a, false, b, (short)0, acc[j], false, false);
    }
  }

  // C/D layout: VGPR r -> (M=r, N=lane) lanes 0-15 ; (M=r+8, N=lane-16) lanes 16-31.
  // Row sum-of-squares: reduce across the 16 lanes of each half (xor 1/2/4/8 stays in-half).
  float ss[8];
#pragma unroll
  for (int r = 0; r < 8; ++r) {
    float p = 0.0f;
#pragma unroll
    for (int j = 0; j < 8; ++j) p += acc[j][r] * acc[j][r];
    p += __shfl_xor(p, 1, 32);
    p += __shfl_xor(p, 2, 32);
    p += __shfl_xor(p, 4, 32);
    p += __shfl_xor(p, 8, 32);
    ss[r] = p;
  }

  const int mbase = base + ((lane < 16) ? 0 : 8);
  const int col   = lane & 15;
#pragma unroll
  for (int r = 0; r < 8; ++r) {
    const int   noderow = mbase + r;
    const float scale   = snorm[noderow] / fmaxf(sqrtf(ss[r]), EPSN); // snorm >= 0
    float* orow = out + (size_t)noderow * DIMS + col;
#pragma unroll
    for (int j = 0; j < 8; ++j) {
      // L2-normalize then relu then graph-norm (scale >= 0 so order commutes)
      orow[j << 4] = fmaxf(acc[j][r], 0.0f) * scale;
    }
  }
}

// ---------------------------------------------------------------------
extern "C" void kernel_launch(void* const* d_in, const int* in_sizes, int n_in,
                              void* d_out, int out_size, void* d_ws, size_t ws_size,
                              hipStream_t stream) {
  const float* h     = (const float*)d_in[0];
  const float* snorm = (const float*)d_in[1];
  const float* W     = (const float*)d_in[2];
  const int*   src   = (const int*)d_in[3];
  const int*   dst   = (const int*)d_in[4];
  float*       out   = (float*)d_out;

  // workspace layout: summed[N*D] | deg[N] | wpack[8*64*64]
  float* summed = (float*)d_ws;
  float* deg    = summed + (size_t)N_NODES * DIMS;
  float* wpack  = deg + N_NODES;

  const int nzero = N_NODES * DIMS + N_NODES;
  zero_kernel<<<(nzero + 255) / 256, 256, 0, stream>>>(summed, nzero);
  repack_w_kernel<<<(8 * 64 * 32 + 255) / 256, 256, 0, stream>>>(W, wpack);
  scatter_kernel<<<(N_EDGES * 32 + 255) / 256, 256, 0, stream>>>(h, src, dst, summed, deg);
  node_apply_kernel<<<(N_TILES + 7) / 8, 256, 0, stream>>>(h, snorm, summed, deg, wpack, out);
}